// PCNLayer_59665685676269
// MI455X (gfx1250) — compile-verified
//
#include <hip/hip_runtime.h>

typedef __attribute__((ext_vector_type(2))) float v2f;
typedef __attribute__((ext_vector_type(4))) float v4f;
typedef __attribute__((ext_vector_type(8))) float v8f;

#define N_NODES_C 2000
#define DDIM 16
#define ENT 8192      // B*T*D = 8*64*16 elements per tree entry
// 512 rows of 16 -> 32 tiles of 16x16 per entry

__global__ void pcn_zero_kernel(v4f* __restrict__ out4, int n_out4,
                                int* __restrict__ counts, int n_counts) {
    int i = blockIdx.x * blockDim.x + threadIdx.x;
    if (i < n_out4)   out4[i] = (v4f){0.f, 0.f, 0.f, 0.f};
    if (i < n_counts) counts[i] = 0;
}

__global__ void pcn_counts_kernel(const int* __restrict__ cn3, int n,
                                  int* __restrict__ counts) {
    int i = blockIdx.x * blockDim.x + threadIdx.x;
    if (i < n) atomicAdd(&counts[cn3[i]], 1);
}

// One wave32 computes one 16x16 output tile of one child entry:
//   D = h_parent_tile(16x16) @ W^T + b   via 4 chained V_WMMA_F32_16X16X4_F32
//   levels 1/2: h_new = D * scale + x[cn]  -> stored to workspace
//   level  3  : atomicAdd(D * scale) into d_out (x[cn3] folded into finalize:
//               mean(fc*scale + x) = sum(fc*scale)/c + x)
__global__ __launch_bounds__(256)
void pcn_level_kernel(const float* __restrict__ x, const float* __restrict__ A,
                      const float* __restrict__ W, const float* __restrict__ bias,
                      const int* __restrict__ pe, const int* __restrict__ pn,
                      const int* __restrict__ cn,
                      const float* __restrict__ hprev,  // null -> parents from x[pn]
                      float* __restrict__ hout,         // used when atomic_out==0
                      float* __restrict__ outsum,       // used when atomic_out==1
                      int n_child, int atomic_out)
{
    const int lane = threadIdx.x & 31;
    const int wid  = (blockIdx.x * 256 + threadIdx.x) >> 5;  // wave-uniform
    const int c    = wid >> 5;   // child-entry index
    const int t    = wid & 31;   // 16-row tile index within entry
    if (c >= n_child) return;    // uniform per wave -> surviving waves keep EXEC all-ones

    const int m     = lane & 15;         // M row (A) / N col (B,C)
    const int khalf = (lane >> 4) << 1;  // 0 or 2: K sub-slice for this half-wave

    const int parent = pn[c];
    const int child  = cn[c];

    // B fragments: B[k][n] = W[n][k] (W is row-major f x d), K slice 4s+khalf
    v2f wf[4];
#pragma unroll
    for (int s = 0; s < 4; ++s)
        wf[s] = *(const v2f*)(W + m * DDIM + 4 * s + khalf);

    // A fragments: rows of the parent's h tile (row-major 512x16)
    const float* hbase = hprev ? (hprev + (size_t)pe[c] * ENT)
                               : (x     + (size_t)parent * ENT);
    const float* hrow  = hbase + (size_t)(t * 16 + m) * DDIM;
    v2f af[4];
#pragma unroll
    for (int s = 0; s < 4; ++s)
        af[s] = *(const v2f*)(hrow + 4 * s + khalf);

    // C init = bias broadcast along rows (N = m is the output-feature index)
    const float bv = bias[m];
    v8f acc;
#pragma unroll
    for (int j = 0; j < 8; ++j) acc[j] = bv;

    // Chain K: D = sum_s A_s x B_s + C
#pragma unroll
    for (int s = 0; s < 4; ++s)
        acc = __builtin_amdgcn_wmma_f32_16x16x4_f32(
                  false, af[s], false, wf[s], (short)0, acc, false, false);

    // scale[c, d=m] = A[m, pn, cn]
    const float scale = A[(size_t)m * N_NODES_C * N_NODES_C
                          + (size_t)parent * N_NODES_C + child];
    const int rhi  = (lane >> 4) << 3;  // +8 rows for upper half-wave
    const int base = (t * 16 + rhi) * DDIM + m;

    if (atomic_out) {
        // leaf segment-sum of fc*scale only; x added in finalize
        float* o = outsum + (size_t)child * ENT + base;
#pragma unroll
        for (int j = 0; j < 8; ++j)
            atomicAdd(&o[j * DDIM], acc[j] * scale);   // global_atomic_add_f32
    } else {
        const float* xe = x + (size_t)child * ENT + base;
        float xv[8];
#pragma unroll
        for (int j = 0; j < 8; ++j) xv[j] = xe[j * DDIM];  // clause of 8 loads
        float* o = hout + (size_t)c * ENT + base;
#pragma unroll
        for (int j = 0; j < 8; ++j)
            o[j * DDIM] = acc[j] * scale + xv[j];
    }
}

// out = (c > 0) ? sums/c + x : x      (x folded out of the leaf sum)
__global__ void pcn_finalize_kernel(const v4f* __restrict__ x4,
                                    const int* __restrict__ counts,
                                    v4f* __restrict__ out4, int n4) {
    int i = blockIdx.x * blockDim.x + threadIdx.x;
    if (i >= n4) return;
    const int node = i >> 11;            // (i*4) / 8192
    const int c = counts[node];
    const v4f xv = x4[i];
    if (c > 0) {
        const float rinv = 1.0f / (float)c;
        v4f s = out4[i];
        s.x = s.x * rinv + xv.x;
        s.y = s.y * rinv + xv.y;
        s.z = s.z * rinv + xv.z;
        s.w = s.w * rinv + xv.w;
        out4[i] = s;
    } else {
        out4[i] = xv;
    }
}

extern "C" void kernel_launch(void* const* d_in, const int* in_sizes, int n_in,
                              void* d_out, int out_size, void* d_ws, size_t ws_size,
                              hipStream_t stream) {
    const float* x  = (const float*)d_in[0];
    const float* A  = (const float*)d_in[1];
    const float* W  = (const float*)d_in[2];
    const float* b  = (const float*)d_in[3];
    // d_in[4] = root_ids (unused: pn1 already holds root node ids)
    const int* pe1 = (const int*)d_in[5];
    const int* pn1 = (const int*)d_in[6];
    const int* cn1 = (const int*)d_in[7];
    const int* pe2 = (const int*)d_in[8];
    const int* pn2 = (const int*)d_in[9];
    const int* cn2 = (const int*)d_in[10];
    const int* pe3 = (const int*)d_in[11];
    const int* pn3 = (const int*)d_in[12];
    const int* cn3 = (const int*)d_in[13];

    const int n1 = in_sizes[7];   // 600
    const int n2 = in_sizes[10];  // 1800
    const int n3 = in_sizes[13];  // 5400

    float* out = (float*)d_out;

    // workspace: h1 (n1 entries) | h2 (n2 entries) | counts (N_NODES ints)
    float* h1     = (float*)d_ws;
    float* h2     = h1 + (size_t)n1 * ENT;
    int*   counts = (int*)(h2 + (size_t)n2 * ENT);

    const int n_out  = out_size;      // 2000 * 8192
    const int n_out4 = n_out / 4;

    pcn_zero_kernel<<<(n_out4 + 255) / 256, 256, 0, stream>>>(
        (v4f*)out, n_out4, counts, N_NODES_C);
    pcn_counts_kernel<<<(n3 + 255) / 256, 256, 0, stream>>>(cn3, n3, counts);

    // Level 1: parents gathered straight from x[pn1] (h0 = x[root_ids])
    pcn_level_kernel<<<n1 * 4, 256, 0, stream>>>(x, A, W, b, pe1, pn1, cn1,
                                                 nullptr, h1, nullptr, n1, 0);
    // Level 2
    pcn_level_kernel<<<n2 * 4, 256, 0, stream>>>(x, A, W, b, pe2, pn2, cn2,
                                                 h1, h2, nullptr, n2, 0);
    // Level 3 fused with leaf segment-sum into d_out (fc*scale only)
    pcn_level_kernel<<<n3 * 4, 256, 0, stream>>>(x, A, W, b, pe3, pn3, cn3,
                                                 h2, nullptr, out, n3, 1);

    pcn_finalize_kernel<<<(n_out4 + 255) / 256, 256, 0, stream>>>(
        (const v4f*)x, counts, (v4f*)out, n_out4);
}